// DetectionLoss_61624190763377
// MI455X (gfx1250) — compile-verified
//
#include <hip/hip_runtime.h>
#include <math.h>

// Problem constants (match reference).
#define B_   512
#define N_   252
#define C_   256
#define NBLK1 504          // B_*N_ / 256 == 129024/256 exactly

// Workspace layout (as floats / ints). Total = (2048 + 129024)*4 = 512 KB.
#define WS_BCE 0           // 504 floats: per-block BCE partial sums
#define WS_MSE 512         // 504 floats: per-block MSE partial sums
#define WS_CE  1024        // 512 floats: per-batch CE partial sums
#define WS_INV 2048        // B_*N_ ints : inverse class map (cls j -> row i, or -1)

typedef __attribute__((ext_vector_type(2))) float v2f;
typedef __attribute__((ext_vector_type(8))) float v8f;

// ---------------------------------------------------------------- K0: init ws
__global__ void k0_init(float* __restrict__ ws) {
    int tid = blockIdx.x * 256 + threadIdx.x;     // 512*256 = 131072 = 2048 + 129024
    if (tid < WS_INV) ws[tid] = 0.0f;
    else              ((int*)ws)[tid] = -1;
}

// ------------------------------------------- K1: BCE partials + class scatter
__global__ void __launch_bounds__(256) k1_bce_scatter(const float* __restrict__ outp,
                                                      const float* __restrict__ tgt,
                                                      float* __restrict__ ws) {
    int tid = blockIdx.x * 256 + threadIdx.x;     // exactly B_*N_ threads
    int b = tid / N_, i = tid - b * N_;
    const float* trow = tgt + ((size_t)b * N_ + i) * C_;
    float t = trow[0];
    float p = outp[((size_t)b * N_ + i) * C_];
    float term = t * logf(p) + (1.0f - t) * log1pf(-p);
    if (t != 0.0f) {
        int cls = (int)trow[4];                   // permutation: unique writer per slot
        ((int*)ws)[WS_INV + b * N_ + cls] = i;
    }
    // deterministic block reduction
    #pragma unroll
    for (int off = 16; off; off >>= 1) term += __shfl_xor(term, off, 32);
    __shared__ float red[8];
    int lane = threadIdx.x & 31, wv = threadIdx.x >> 5;
    if (lane == 0) red[wv] = term;
    __syncthreads();
    if (threadIdx.x == 0) {
        float s = 0.0f;
        #pragma unroll
        for (int w = 0; w < 8; ++w) s += red[w];
        ws[WS_BCE + blockIdx.x] = s;
    }
}

// ------------------------------------------------------- K2: MSE partials
__global__ void __launch_bounds__(256) k2_mse(const float* __restrict__ outp,
                                              const float* __restrict__ tgt,
                                              float* __restrict__ ws) {
    int tid = blockIdx.x * 256 + threadIdx.x;
    int b = tid / N_, j = tid - b * N_;
    const float* orow = outp + ((size_t)b * N_ + j) * C_;
    float mj = (tgt[((size_t)b * N_ + j) * C_] != 0.0f) ? 1.0f : 0.0f;
    float f1 = mj * orow[1], f2 = mj * orow[2], f3 = mj * orow[3];
    int is = ((const int*)ws)[WS_INV + b * N_ + j];
    float s1 = 0.0f, s2 = 0.0f, s3 = 0.0f;
    if (is >= 0) {
        const float* tr = tgt + ((size_t)b * N_ + is) * C_;
        s1 = tr[1]; s2 = tr[2]; s3 = tr[3];
    }
    float dw = sqrtf(f3) - sqrtf(s3);
    float term = (f1 - s1) * (f1 - s1) + (f2 - s2) * (f2 - s2) + 2.0f * dw * dw;
    #pragma unroll
    for (int off = 16; off; off >>= 1) term += __shfl_xor(term, off, 32);
    __shared__ float red[8];
    int lane = threadIdx.x & 31, wv = threadIdx.x >> 5;
    if (lane == 0) red[wv] = term;
    __syncthreads();
    if (threadIdx.x == 0) {
        float s = 0.0f;
        #pragma unroll
        for (int w = 0; w < 8; ++w) s += red[w];
        ws[WS_MSE + blockIdx.x] = s;
    }
}

// ---- K3: CE via f32 WMMA column-sum of exp(logits). 1 block per batch,
// ---- 8 waves x 32-channel stripes, A = ones(16x4), B = 4x16 exp tile.
__global__ void __launch_bounds__(256) k3_ce(const float* __restrict__ outp,
                                             const float* __restrict__ tgt,
                                             float* __restrict__ ws) {
    __shared__ float maskv[N_];
    __shared__ float red[8];
    int b = blockIdx.x;
    int t = threadIdx.x;
    if (t < N_)
        maskv[t] = (tgt[((size_t)b * N_ + t) * C_] != 0.0f) ? 1.0f : 0.0f;
    __syncthreads();

    int lane = t & 31, wv = t >> 5;
    int ch0  = wv * 32;                       // this wave's channel stripe
    int c0   = 4 + ch0 + (lane & 15);         // tile0 column (<= 243, always in range)
    int c1   = c0 + 16;  if (c1 > C_ - 1) c1 = C_ - 1;   // wave7 tail: clamp, discard later
    const float* base = outp + (size_t)b * N_ * C_;
    int rbase = (lane < 16) ? 0 : 2;          // K-slot split across lane halves

    v2f ones = {1.0f, 1.0f};                  // A = ones(16x4): layout-invariant
    v8f acc0 = {};                            // C accumulators, full f32
    v8f acc1 = {};

    for (int i0 = 0; i0 < N_; i0 += 4) {      // 63 chunks of 4 rows
        int rA = i0 + rbase, rB = rA + 1;
        if (i0 + 8 < N_)                      // keep the HBM stream ahead
            __builtin_prefetch(&base[(size_t)(rA + 8) * C_ + c0], 0, 1);
        float mA = maskv[rA], mB = maskv[rB];
        float x0a = base[(size_t)rA * C_ + c0] * mA;   // filtered logit (0 if masked out)
        float x0b = base[(size_t)rB * C_ + c0] * mB;
        float x1a = base[(size_t)rA * C_ + c1] * mA;
        float x1b = base[(size_t)rB * C_ + c1] * mB;
        v2f b0, b1;
        b0.x = __expf(x0a); b0.y = __expf(x0b);        // exp(0)=1 for masked rows (matches ref)
        b1.x = __expf(x1a); b1.y = __expf(x1b);
        acc0 = __builtin_amdgcn_wmma_f32_16x16x4_f32(false, ones, false, b0,
                                                     (short)0, acc0, false, false);
        acc1 = __builtin_amdgcn_wmma_f32_16x16x4_f32(false, ones, false, b1,
                                                     (short)0, acc1, false, false);
    }

    // D[m,n] = sum_i exp(v_i) for column n; VGPR0 holds it for lane%16 == n.
    float s = (lane < 16) ? acc0[0] : acc1[0];
    int k = ch0 + lane;                       // column index 0..255 (valid < 252)
    float term = 0.0f;
    if (k < N_) {
        int invk = ((const int*)ws)[WS_INV + b * N_ + k];
        int r = (invk >= 0) ? k : 0;          // tgt_cls[b,k]
        float vr = maskv[r] * base[(size_t)r * C_ + 4 + k];
        term = logf(s) - vr;                  // = -picked[b,k]; stable: s in [252, 685]
    }
    #pragma unroll
    for (int off = 16; off; off >>= 1) term += __shfl_xor(term, off, 32);
    if (lane == 0) red[wv] = term;
    __syncthreads();
    if (t == 0) {
        float sm = 0.0f;
        #pragma unroll
        for (int w = 0; w < 8; ++w) sm += red[w];
        ws[WS_CE + b] = sm;
    }
}

// ---------------------------------------------------------- K4: final combine
__global__ void k4_final(const float* __restrict__ ws, float* __restrict__ res) {
    if (threadIdx.x != 0 || blockIdx.x != 0) return;
    float bce_s = 0.0f, mse_s = 0.0f, ce_s = 0.0f;
    for (int i = 0; i < NBLK1; ++i) bce_s += ws[WS_BCE + i];
    for (int i = 0; i < NBLK1; ++i) mse_s += ws[WS_MSE + i];
    for (int i = 0; i < B_;    ++i) ce_s  += ws[WS_CE  + i];
    const float inv_bn = 1.0f / (float)(B_ * N_);
    float bce = -bce_s * inv_bn;
    float mse =  mse_s * inv_bn;              // = Lx + Ly + 2*Lwh
    float ce  =  ce_s  * inv_bn;              // mean over B*252 (== B*N here)
    // 10*mse + bce + 0.5*(1-bce) + ce
    res[0] = 10.0f * mse + 0.5f * bce + 0.5f + ce;
}

extern "C" void kernel_launch(void* const* d_in, const int* in_sizes, int n_in,
                              void* d_out, int out_size, void* d_ws, size_t ws_size,
                              hipStream_t stream) {
    const float* outp = (const float*)d_in[0];   // 'output' (B,N,C) f32
    const float* tgt  = (const float*)d_in[1];   // 'target' (B,N,C) f32
    float* ws  = (float*)d_ws;                   // needs 512 KB
    float* res = (float*)d_out;

    k0_init       <<<512,   256, 0, stream>>>(ws);
    k1_bce_scatter<<<NBLK1, 256, 0, stream>>>(outp, tgt, ws);
    k2_mse        <<<NBLK1, 256, 0, stream>>>(outp, tgt, ws);
    k3_ce         <<<B_,    256, 0, stream>>>(outp, tgt, ws);
    k4_final      <<<1,     1,   0, stream>>>(ws, res);
}